// SAGE_51170240364826
// MI455X (gfx1250) — compile-verified
//
#include <hip/hip_runtime.h>

// ---------------------------------------------------------------------------
// GraphSAGE 2-layer inference for MI455X (gfx1250), fp32 via V_WMMA_F32_16X16X4_F32
// ---------------------------------------------------------------------------

typedef __attribute__((ext_vector_type(2))) float v2f;
typedef __attribute__((ext_vector_type(8))) float v8f;

#define NNODES 50000
#define CIN    128   // K dimension for both layers (in_c == hid_c == 128)

// ---- edge scatter: s[dst] += feat[src]; cnt[dst] += 1 (first layer only) ----
// 128 lanes per edge -> fully coalesced 512B gather + 512B atomic scatter.
__global__ void sage_scatter(const float* __restrict__ feat,
                             const int*   __restrict__ src,
                             const int*   __restrict__ dst,
                             float* __restrict__ s,
                             float* __restrict__ cnt,
                             int E) {
    int e = blockIdx.x * 2 + (threadIdx.x >> 7);   // 256 threads = 2 edges/block
    if (e >= E) return;
    int c  = threadIdx.x & 127;
    int sv = src[e];
    int dv = dst[e];
    atomicAdd(&s[dv * CIN + c], feat[sv * CIN + c]);
    if (cnt != nullptr && c == 0) atomicAdd(&cnt[dv], 1.0f);
}

// ---- mean normalize: s[i] /= max(cnt[row],1) ----
__global__ void sage_normalize(float* __restrict__ s, const float* __restrict__ cnt) {
    int i = blockIdx.x * blockDim.x + threadIdx.x;
    if (i < NNODES * CIN) {
        s[i] = s[i] / fmaxf(cnt[i >> 7], 1.0f);
    }
}

// ---- fused dual GEMM:  out = Aagg @ Wl^T + Ax @ Wr^T + b  (optional relu) ----
// One wave computes a 16x16 output tile, alternating two f32 WMMAs per K-step
// into a single v8f accumulator. EXEC is all-ones by construction (no guards),
// M = 50000 = 3125*16, N in {128,64}, K = 128: all exact multiples.
__global__ void sage_gemm(const float* __restrict__ Aagg, const float* __restrict__ Ax,
                          const float* __restrict__ Wl,   const float* __restrict__ Wr,
                          const float* __restrict__ bias, float* __restrict__ out,
                          int Nout, int relu) {
    const int lane = threadIdx.x & 31;
    const int wave = threadIdx.x >> 5;            // 4 waves/block -> 4 N-tiles
    const int m0   = blockIdx.y * 16;
    const int n0   = (blockIdx.x * 4 + wave) * 16;
    const int half = lane >> 4;                   // 0: K pair {k,k+1}; 1: {k+2,k+3}
    const int l15  = lane & 15;
    const int row  = m0 + l15;                    // A fragment row (M)
    const int col  = n0 + l15;                    // B fragment col (N)
    const int kh   = half * 2;

    // A: [M,K] row-major; fragment = contiguous 64-bit load A[row][k+kh .. +1]
    const float* aL = Aagg + row * CIN + kh;
    const float* aR = Ax   + row * CIN + kh;
    // B = W^T (W is [Nout,K] row-major): B[k][col] = W[col][k] -> contiguous load
    const float* bL = Wl + col * CIN + kh;
    const float* bR = Wr + col * CIN + kh;

    v8f c = {};
#pragma unroll
    for (int k = 0; k < CIN; k += 4) {
        v2f a0 = *(const v2f*)(aL + k);
        v2f b0 = *(const v2f*)(bL + k);
        c = __builtin_amdgcn_wmma_f32_16x16x4_f32(false, a0, false, b0,
                                                  (short)0, c, false, false);
        v2f a1 = *(const v2f*)(aR + k);
        v2f b1 = *(const v2f*)(bR + k);
        c = __builtin_amdgcn_wmma_f32_16x16x4_f32(false, a1, false, b1,
                                                  (short)0, c, false, false);
    }

    // C/D layout: lanes 0-15 -> M = m0+vgpr, N = col; lanes 16-31 -> M = m0+8+vgpr
    const float bv = bias[col];
    const int   mb = m0 + half * 8;
#pragma unroll
    for (int i = 0; i < 8; ++i) {
        float v = c[i] + bv;
        if (relu) v = fmaxf(v, 0.0f);
        out[(mb + i) * Nout + col] = v;
    }
}

extern "C" void kernel_launch(void* const* d_in, const int* in_sizes, int n_in,
                              void* d_out, int out_size, void* d_ws, size_t ws_size,
                              hipStream_t stream) {
    // setup_inputs order: x, edge_index, W1l, b1l, W1r, W2l, b2l, W2r
    const float* x   = (const float*)d_in[0];
    const int*   ei  = (const int*)d_in[1];   // per harness doc: integer -> const int*
    const float* W1l = (const float*)d_in[2];
    const float* b1l = (const float*)d_in[3];
    const float* W1r = (const float*)d_in[4];
    const float* W2l = (const float*)d_in[5];
    const float* b2l = (const float*)d_in[6];
    const float* W2r = (const float*)d_in[7];
    float*       out = (float*)d_out;

    const int  E   = in_sizes[1] / 2;         // edge_index is [2, E]
    const int* src = ei;
    const int* dst = ei + E;

    // workspace: cnt (padded) | s (agg sums) | h (hidden activations)
    float* cnt = (float*)d_ws;
    float* s   = cnt + 50176;                 // 256-aligned pad of NNODES
    float* h   = s + (size_t)NNODES * CIN;

    const size_t sBytes = (size_t)NNODES * CIN * sizeof(float);
    hipMemsetAsync(cnt, 0, 50176 * sizeof(float), stream);
    hipMemsetAsync(s,   0, sBytes, stream);

    const int sgrid = (E + 1) / 2;
    const int ntot  = NNODES * CIN;

    // ---- layer 1 ----
    sage_scatter  <<<sgrid, 256, 0, stream>>>(x, src, dst, s, cnt, E);
    sage_normalize<<<(ntot + 255) / 256, 256, 0, stream>>>(s, cnt);
    sage_gemm     <<<dim3(2, NNODES / 16), 128, 0, stream>>>(s, x, W1l, W1r, b1l,
                                                             h, 128, /*relu=*/1);
    // ---- layer 2 ----
    hipMemsetAsync(s, 0, sBytes, stream);
    sage_scatter  <<<sgrid, 256, 0, stream>>>(h, src, dst, s, nullptr, E);
    sage_normalize<<<(ntot + 255) / 256, 256, 0, stream>>>(s, cnt);
    sage_gemm     <<<dim3(1, NNODES / 16), 128, 0, stream>>>(s, h, W2l, W2r, b2l,
                                                             out, 64, /*relu=*/0);
}